// loop_model_24507083391507
// MI455X (gfx1250) — compile-verified
//
#include <hip/hip_runtime.h>
#include <stdint.h>

// out[i] = (((x[i]*w + b)*w + b) ... y times)   -- elementwise affine stream.
// HBM-bound: 256 MB traffic => ~11 us floor at 23.3 TB/s. No WMMA applicable
// (feature dim is 1). Use the gfx1250 async-to-LDS path (ASYNCcnt) with
// per-wave double buffering + B128 accesses + non-temporal stores.
// K and stride are computed on the host to avoid a 64-bit software divide
// in the kernel prologue (seen as ~300 SALU ops in round 2).

typedef float v4f __attribute__((ext_vector_type(4)));
typedef int   v4i __attribute__((ext_vector_type(4)));

#define THREADS 256
#define NWAVES (THREADS / 32)
#define MAXBLOCKS 4096

#if defined(__HIP_DEVICE_COMPILE__) && defined(__gfx1250__) && \
    __has_builtin(__builtin_amdgcn_global_load_async_to_lds_b128)
#define USE_ASYNC_LDS 1
#else
#define USE_ASYNC_LDS 0
#endif

#if USE_ASYNC_LDS
// Builtin signature (clang-verified): (int4 AS1* src, int4 AS3* dst, Imm, Imm)
typedef __attribute__((address_space(1))) v4i gv4i_t;  // global int4
typedef __attribute__((address_space(3))) v4i lv4i_t;  // LDS int4
#endif

__device__ __forceinline__ v4f apply_affine(v4f v, float w, float bb, int y) {
  // Match reference semantics: y sequential (mul, add) steps in fp32.
  for (int k = 0; k < y; ++k) {
    v.x = __fadd_rn(__fmul_rn(v.x, w), bb);
    v.y = __fadd_rn(__fmul_rn(v.y, w), bb);
    v.z = __fadd_rn(__fmul_rn(v.z, w), bb);
    v.w = __fadd_rn(__fmul_rn(v.w, w), bb);
  }
  return v;
}

__global__ __launch_bounds__(THREADS) void affine_stream_kernel(
    const float* __restrict__ x, const float* __restrict__ Wp,
    const float* __restrict__ bp, const int* __restrict__ yp,
    float* __restrict__ out, long long n, long long n4, long long K,
    long long stride) {
  const float w  = Wp[0];
  const float bb = bp[0];
  const int   y  = yp[0];

  const long long tid = (long long)blockIdx.x * THREADS + threadIdx.x;

  const v4f* __restrict__ x4 = (const v4f*)x;
  v4f* __restrict__ o4       = (v4f*)out;

#if USE_ASYNC_LDS
  // Per-wave private double buffer: no barriers needed; each wave reads only
  // the LDS region it async-loaded itself, gated by s_wait_asynccnt.
  __shared__ v4f stage[NWAVES][2][32];
  const int wave = threadIdx.x >> 5;
  const int lane = threadIdx.x & 31;

  if (K > 0) {
    // Prologue: async-load tile 0 into buffer 0 (16 B/lane, ASYNCcnt++).
    __builtin_amdgcn_global_load_async_to_lds_b128(
        (gv4i_t*)(x4 + tid), (lv4i_t*)&stage[wave][0][lane], 0, 0);
    int buf = 0;
    for (long long k = 0; k < K; ++k) {
      const long long i = tid + k * stride;
      if (k + 1 < K) {
        // WAR guard: our ds_load of buf^1 (iteration k-1) must retire before
        // the async engine overwrites it (async vs DS are unordered).
        asm volatile("s_wait_dscnt 0" ::: "memory");
        __builtin_amdgcn_global_load_async_to_lds_b128(
            (gv4i_t*)(x4 + i + stride), (lv4i_t*)&stage[wave][buf ^ 1][lane],
            0, 0);
        // Leave only the just-issued prefetch outstanding.
        asm volatile("s_wait_asynccnt 1" ::: "memory");
      } else {
        asm volatile("s_wait_asynccnt 0" ::: "memory");
      }
      v4f v = stage[wave][buf][lane];  // ds_load_b128
      v = apply_affine(v, w, bb, y);
      // Output is written once and never re-read: NT store, keep L2 for input.
      __builtin_nontemporal_store(v, o4 + i);
      buf ^= 1;
    }
  }
#else
  for (long long k = 0; k < K; ++k) {
    const long long i = tid + k * stride;
    v4f v = __builtin_nontemporal_load(x4 + i);
    v = apply_affine(v, w, bb, y);
    __builtin_nontemporal_store(v, o4 + i);
  }
#endif

  // Remainder float4s (n4 % stride) — direct path, possibly divergent.
  for (long long i = K * stride + tid; i < n4; i += stride) {
    v4f v = x4[i];
    v = apply_affine(v, w, bb, y);
    o4[i] = v;
  }
  // Scalar tail (n % 4) — dead for N = 2^25.
  for (long long i = (n4 << 2) + tid; i < n; i += stride) {
    float v = x[i];
    for (int k = 0; k < y; ++k) v = __fadd_rn(__fmul_rn(v, w), bb);
    out[i] = v;
  }
}

extern "C" void kernel_launch(void* const* d_in, const int* in_sizes, int n_in,
                              void* d_out, int out_size, void* d_ws,
                              size_t ws_size, hipStream_t stream) {
  const float* x  = (const float*)d_in[0];
  const float* W  = (const float*)d_in[1];
  const float* b  = (const float*)d_in[2];
  const int*   y  = (const int*)d_in[3];
  float* out      = (float*)d_out;

  const long long n  = (long long)in_sizes[0];
  const long long n4 = n >> 2;
  long long want = (n4 + THREADS - 1) / THREADS;
  if (want < 1) want = 1;
  if (want > MAXBLOCKS) want = MAXBLOCKS;
  const int blocks = (int)want;

  const long long stride = (long long)blocks * THREADS;
  const long long K      = n4 / stride;  // host-side divide, not in-kernel

  affine_stream_kernel<<<blocks, THREADS, 0, stream>>>(x, W, b, y, out, n, n4,
                                                       K, stride);
}